// TargetOpinionPairRepresentation_47742856462633
// MI455X (gfx1250) — compile-verified
//
#include <hip/hip_runtime.h>
#include <stdint.h>

// Problem constants (fixed by the reference's setup_inputs).
constexpr int kB = 16, kS = 4096, kD = 256, kT = 128, kO = 128, kE = 128;
constexpr int kW = 2 * kD + kE;   // 640 floats per output row

typedef float v4f __attribute__((ext_vector_type(4)));

// Low 32 bits of a generic pointer to __shared__ data are the LDS byte address
// (generic LDS ptr = {SHARED_BASE aperture, offset}; the LDS address is the low 32 bits).
__device__ __forceinline__ uint32_t lds_off(const void* p) {
  return (uint32_t)(uintptr_t)p;
}

// CDNA5 async global->LDS DMA (ASYNCcnt-tracked), GV addressing mode:
//   global_load_async_to_lds_b128 vdst(LDS addr), vaddr(64b), off
__device__ __forceinline__ void async_g2l_b128(uint32_t lds_addr, const void* gptr) {
  uint64_t ga = (uint64_t)(uintptr_t)gptr;
  asm volatile("global_load_async_to_lds_b128 %0, %1, off"
               :: "v"(lds_addr), "v"(ga)
               : "memory");
}

__device__ __forceinline__ void wait_async0() {
  asm volatile("s_wait_asynccnt 0" ::: "memory");
}

__global__ __launch_bounds__(256)
void pair_rep_kernel(const float* __restrict__ spans,            // [B,S,D]
                     const float* __restrict__ dist_emb,         // [10,E]
                     const int*   __restrict__ span_indices,     // [S,2]
                     const int*   __restrict__ target_indices,   // [B,T]
                     const int*   __restrict__ opinion_indices,  // [B,O]
                     float*       __restrict__ out)              // [B,T*O,640]
{
  __shared__ __align__(16) float s_tgt[kD];        // 1 KB: target span row
  __shared__ __align__(16) float s_demb[10 * kE];  // 5 KB: full bucket-embedding table
  __shared__ int s_pack[kO];                       // (bucket<<12) | opinion_span_index

  const int tid = threadIdx.x;
  const int bt  = blockIdx.x;        // bt = b*T + t   (T == 128)
  const int b   = bt >> 7;

  // Target bounds (wave-uniform scalar loads).
  const int ti = target_indices[bt];
  const int ta = span_indices[2 * ti + 0];
  const int tb = span_indices[2 * ti + 1];

  // Opinion gather + bucket compute (branchless searchsorted-right minus 1),
  // packed into one LDS word per opinion: oi in [0,4096) -> 12 bits, bucket -> 4 bits.
  if (tid < kO) {
    const int oi = opinion_indices[b * kO + tid];
    const int c  = span_indices[2 * oi + 0];
    const int d  = span_indices[2 * oi + 1];
    int d0 = tb - c; if (d0 < 0) d0 = -d0;
    int d1 = ta - d; if (d1 < 0) d1 = -d1;
    const int dist = d0 < d1 ? d0 : d1;
    const int bk = (dist >= 1) + (dist >= 2) + (dist >= 3) + (dist >= 4) +
                   (dist >= 5) + (dist >= 8) + (dist >= 16) + (dist >= 32) +
                   (dist >= 64);
    s_pack[tid] = oi | (bk << 12);
  }

  // Async-stage dist_emb (320 x b128) and the target row (64 x b128) into LDS.
  // All issues happen with full-wave EXEC (tid<64 == waves 0..1 fully active).
  {
    const char*    dsrc = (const char*)dist_emb;
    const uint32_t ddst = lds_off(s_demb);
    async_g2l_b128(ddst + 16u * (uint32_t)tid, dsrc + 16u * (uint32_t)tid);
    if (tid < 64) {
      async_g2l_b128(ddst + 16u * (uint32_t)(tid + 256), dsrc + 16u * (uint32_t)(tid + 256));
      const char* tsrc = (const char*)(spans + ((size_t)b * kS + (size_t)ti) * kD);
      async_g2l_b128(lds_off(s_tgt) + 16u * (uint32_t)tid, tsrc + 16u * (uint32_t)tid);
    }
  }
  wait_async0();       // this wave's async copies have landed in LDS
  __syncthreads();     // everyone's have

  // Each wave owns opinions o = wave, wave+8, ... (16 rows per wave).
  const int  wave = tid >> 5;
  const int  lane = tid & 31;
  const v4f* s_tgt4 = (const v4f*)s_tgt;

  // Target half is invariant across o: hoist into registers (2 x b128 LDS reads total).
  const v4f tv0 = s_tgt4[lane];
  const v4f tv1 = s_tgt4[lane + 32];

  // Base pointers so the inner loop is pure 32-bit shift/mask address math.
  const char* spansB = (const char*)(spans + (size_t)b * kS * kD);  // batch b's rows
  const char* dembB  = (const char*)s_demb;
  v4f* orow = (v4f*)(out + ((size_t)bt * kO + (size_t)wave) * kW);  // row 'wave'
  constexpr size_t kRowStride4 = (size_t)8 * kW / 4;                // 8 rows of v4f

#pragma unroll 4
  for (int o = wave; o < kO; o += 8, orow += kRowStride4) {
    const int p  = s_pack[o];                                  // one ds_load_b32
    const v4f* opn = (const v4f*)(spansB + ((uint32_t)(p & 0xFFF) << 10)); // oi*1024B
    const v4f* bem = (const v4f*)(dembB + ((uint32_t)(p >> 12) << 9));     // bk*512B

    // Prefetch next opinion row (1 KB = 8 cachelines); rows are reused by the
    // other t-blocks of this batch, so keep them temporal in L2.
    const int on = o + 8;
    if (on < kO) {
      const char* np = spansB + ((uint32_t)(s_pack[on] & 0xFFF) << 10);
      __builtin_prefetch(np + 32 * lane, 0, 3);
    }

    const v4f a2 = opn[lane];        // opinion half: global b128 (L2 hit)
    const v4f a3 = opn[lane + 32];
    const v4f a4 = bem[lane];        // embedding: LDS b128

    // Streaming (non-temporal) b128 stores: output is never re-read; keep L2
    // for the reused spans/dist_emb working set.
    __builtin_nontemporal_store(tv0, &orow[lane]);
    __builtin_nontemporal_store(tv1, &orow[lane + 32]);
    __builtin_nontemporal_store(a2,  &orow[lane + 64]);
    __builtin_nontemporal_store(a3,  &orow[lane + 96]);
    __builtin_nontemporal_store(a4,  &orow[lane + 128]);
  }
}

extern "C" void kernel_launch(void* const* d_in, const int* in_sizes, int n_in,
                              void* d_out, int out_size, void* d_ws, size_t ws_size,
                              hipStream_t stream) {
  (void)in_sizes; (void)n_in; (void)out_size; (void)d_ws; (void)ws_size;
  const float* spans           = (const float*)d_in[0];
  const float* dist_emb        = (const float*)d_in[1];
  const int*   span_indices    = (const int*)d_in[2];
  const int*   target_indices  = (const int*)d_in[3];
  const int*   opinion_indices = (const int*)d_in[4];
  float*       out             = (float*)d_out;

  dim3 grid(kB * kT);   // 2048 blocks: one per (b, t)
  dim3 block(256);      // 8 wave32s
  pair_rep_kernel<<<grid, block, 0, stream>>>(spans, dist_emb, span_indices,
                                              target_indices, opinion_indices, out);
}